// triplet_to_edge_layer_36189394436678
// MI455X (gfx1250) — compile-verified
//
#include <hip/hip_runtime.h>
#include <hip/hip_bf16.h>

// ---------------------------------------------------------------------------
// triplet_to_edge_layer for MI455X (gfx1250, wave32)
//
// Pipeline (all on `stream`):
//   k_prep      : Wc[64x8] = W_sbf(64x7) @ W_sbf0(7x7)   (folds the 2 sbf projections)
//   k_zero      : agg = 0
//   k_gemm_down : X = silu(edge_attr @ W_down^T)          [E,64]   (f32 WMMA)
//   k_triplet   : agg[idx_ji] += X[idx_kj] * (sbf @ Wc^T) (f32 atomics, L2-resident)
//   k_up_e1     : u = silu(agg @ W_up^T) kept in LDS, then
//                 out = silu([edge_attr,u] @ W_e1^T + b) + edge_attr  (f32 WMMA, fused)
// ---------------------------------------------------------------------------

#define E_EDGES   262144
#define T_TRIP    2097152
#define HIDDEN    128
#define INT_EMB   64
#define NUM_SPH   7
#define MT_TILES  (E_EDGES / 16)   // 16384 M-tiles of 16 rows

typedef __attribute__((ext_vector_type(2))) float v2f;
typedef __attribute__((ext_vector_type(8))) float v8f;

// Fast silu: v_exp_f32 + v_rcp_f32 instead of the IEEE div ladder.
__device__ __forceinline__ float silu(float v) {
    return v * __builtin_amdgcn_rcpf(1.0f + __expf(-v));
}

// D(16x16,f32) += A(16x4,f32) x B(4x16,f32) ; one wave32 owns the whole tile.
__device__ __forceinline__ v8f wmma_f32_k4(v2f a, v2f b, v8f c) {
#if __has_builtin(__builtin_amdgcn_wmma_f32_16x16x4_f32)
    // (neg_a, A, neg_b, B, c_mod, C, reuse_a, reuse_b)
    return __builtin_amdgcn_wmma_f32_16x16x4_f32(false, a, false, b, (short)0, c,
                                                 false, false);
#else
    asm("v_wmma_f32_16x16x4_f32 %0, %1, %2, %0" : "+v"(c) : "v"(a), "v"(b));
    return c;
#endif
}

// ---------------------------------------------------------------------------
// k_prep: Wc[j][k] = sum_q W_sbf[j][q] * W_sbf0[q][k], padded to [64][8]
// ---------------------------------------------------------------------------
__global__ void k_prep(const float* __restrict__ W_sbf0,
                       const float* __restrict__ W_sbf,
                       float* __restrict__ Wc) {
    int j = threadIdx.x >> 3;   // 0..63
    int k = threadIdx.x & 7;    // 0..7
    float v = 0.0f;
    if (k < NUM_SPH) {
        #pragma unroll
        for (int q = 0; q < NUM_SPH; ++q)
            v = fmaf(W_sbf[j * NUM_SPH + q], W_sbf0[q * NUM_SPH + k], v);
    }
    Wc[j * 8 + k] = v;
}

__global__ void k_zero(float4* __restrict__ p, int n4) {
    int i = blockIdx.x * blockDim.x + threadIdx.x;
    if (i < n4) p[i] = make_float4(0.f, 0.f, 0.f, 0.f);
}

// ---------------------------------------------------------------------------
// k_gemm_down: X[e][j] = silu( sum_k edge_attr[e][k] * W_down[j][k] )
//   One wave per 16x16 tile; 8 waves / block; 4 N-tiles (INT_EMB=64).
// Fragment layout (ISA 7.12.2, f32 16x16x4):
//   A: lane<16 -> M=lane, {K+0,K+1}; lane>=16 -> M=lane-16, {K+2,K+3}
//   B: lane<16 -> N=lane, {K+0,K+1}; lane>=16 -> N=lane-16, {K+2,K+3}
//   D: vgpr r  -> M = r + 8*(lane>=16), N = lane&15
// ---------------------------------------------------------------------------
__global__ void __launch_bounds__(256)
k_gemm_down(const float* __restrict__ A, const float* __restrict__ W,
            float* __restrict__ X) {
    const int wave = threadIdx.x >> 5;
    const int lane = threadIdx.x & 31;
    const int tile = blockIdx.x * 8 + wave;
    const int mt = tile >> 2;          // 0..MT_TILES-1 (4 waves share one mt)
    const int nt = tile & 3;           // 0..3
    const int r16 = lane & 15;
    const int khi = lane >> 4;         // 0/1 -> K offset 0/2

    const float* arow = A + (size_t)(mt * 16 + r16) * HIDDEN + 2 * khi;
    const float* brow = W + (size_t)(nt * 16 + r16) * HIDDEN + 2 * khi;

    v8f acc = {};
    #pragma unroll 8
    for (int k = 0; k < HIDDEN; k += 4) {
        v2f a = *(const v2f*)(arow + k);
        v2f b = *(const v2f*)(brow + k);
        acc = wmma_f32_k4(a, b, acc);
    }

    const int n  = nt * 16 + r16;
    const int m0 = mt * 16 + 8 * khi;
    #pragma unroll
    for (int r = 0; r < 8; ++r)
        X[(size_t)(m0 + r) * INT_EMB + n] = silu(acc[r]);
}

// ---------------------------------------------------------------------------
// k_triplet: one wave per triplet, 2 columns per lane.
//   s[j]   = sum_q sbf[t][q] * Wc[j][q]
//   agg[idx_ji[t]][j] += X[idx_kj[t]][j] * s[j]   (native f32 atomics in L2)
// ---------------------------------------------------------------------------
__global__ void __launch_bounds__(256)
k_triplet(const float* __restrict__ sbf, const int* __restrict__ idx_kj,
          const int* __restrict__ idx_ji, const float* __restrict__ Wc,
          const float* __restrict__ X, float* __restrict__ agg) {
    __shared__ float lwc[64 * 9];      // stride-9 padding: conflict-free banks
    for (int i = threadIdx.x; i < 64 * 8; i += blockDim.x)
        lwc[(i >> 3) * 9 + (i & 7)] = Wc[i];
    __syncthreads();

    const int wave = threadIdx.x >> 5;
    const int lane = threadIdx.x & 31;
    const int t = blockIdx.x * 8 + wave;       // T is a multiple of 8 waves

    float sb[NUM_SPH];
    const float* srow = sbf + (size_t)t * NUM_SPH;   // broadcast across wave
    #pragma unroll
    for (int q = 0; q < NUM_SPH; ++q) sb[q] = srow[q];

    const int kj = idx_kj[t];
    const int ji = idx_ji[t];
    const float2 xv = *(const float2*)(X + (size_t)kj * INT_EMB + 2 * lane);

    const float* w0 = lwc + (2 * lane) * 9;
    const float* w1 = w0 + 9;
    float s0 = 0.f, s1 = 0.f;
    #pragma unroll
    for (int q = 0; q < NUM_SPH; ++q) {
        s0 = fmaf(sb[q], w0[q], s0);
        s1 = fmaf(sb[q], w1[q], s1);
    }

    float* dst = agg + (size_t)ji * INT_EMB + 2 * lane;
    __hip_atomic_fetch_add(dst,     xv.x * s0, __ATOMIC_RELAXED, __HIP_MEMORY_SCOPE_AGENT);
    __hip_atomic_fetch_add(dst + 1, xv.y * s1, __ATOMIC_RELAXED, __HIP_MEMORY_SCOPE_AGENT);
}

// ---------------------------------------------------------------------------
// k_up_e1: fused up-projection + edge update (one 16-row M-tile per block).
//   Phase 1: each of 8 waves computes one 16x16 tile of
//              u = silu(agg @ W_up^T)            (K=64, f32 WMMA)
//            and stores it to LDS (stride-130 padded: conflict-free).
//   Phase 2: each wave computes one 16x16 tile of
//              out = silu([edge_attr,u] @ W_e1^T + b) + edge_attr
//            K=256 as two 128-phases: edge_attr from global, u from LDS.
// Saves the 256 MB U round-trip through HBM.
// ---------------------------------------------------------------------------
#define LDS_STRIDE 130

__global__ void __launch_bounds__(256)
k_up_e1(const float* __restrict__ AGG, const float* __restrict__ W_up,
        const float* __restrict__ EA,  const float* __restrict__ W_e1,
        const float* __restrict__ bias, float* __restrict__ out) {
    __shared__ float lu[16 * LDS_STRIDE];    // 16 x 128 silu(u) tile, padded

    const int wave = threadIdx.x >> 5;       // 0..7 == N-tile
    const int lane = threadIdx.x & 31;
    const int mt = blockIdx.x;               // one M-tile per block
    const int nt = wave;
    const int r16 = lane & 15;
    const int khi = lane >> 4;

    // ---- phase 1: u tile = silu(agg @ W_up^T) -> LDS
    {
        const float* arow = AGG  + (size_t)(mt * 16 + r16) * INT_EMB + 2 * khi;
        const float* brow = W_up + (size_t)(nt * 16 + r16) * INT_EMB + 2 * khi;
        v8f acc = {};
        #pragma unroll
        for (int k = 0; k < INT_EMB; k += 4) {
            v2f a = *(const v2f*)(arow + k);
            v2f b = *(const v2f*)(brow + k);
            acc = wmma_f32_k4(a, b, acc);
        }
        const int n = nt * 16 + r16;
        #pragma unroll
        for (int r = 0; r < 8; ++r)
            lu[(8 * khi + r) * LDS_STRIDE + n] = silu(acc[r]);
    }
    __syncthreads();

    // ---- phase 2: out tile = silu([EA,u] @ W_e1^T + b) + EA
    const float* arow0 = EA   + (size_t)(mt * 16 + r16) * HIDDEN + 2 * khi;
    const float* brow  = W_e1 + (size_t)(nt * 16 + r16) * (2 * HIDDEN) + 2 * khi;
    const float* lrow  = lu + r16 * LDS_STRIDE + 2 * khi;

    v8f acc = {};
    #pragma unroll 8
    for (int k = 0; k < HIDDEN; k += 4) {          // K phase 1: edge_attr
        v2f a = *(const v2f*)(arow0 + k);
        v2f b = *(const v2f*)(brow + k);
        acc = wmma_f32_k4(a, b, acc);
    }
    #pragma unroll 8
    for (int k = 0; k < HIDDEN; k += 4) {          // K phase 2: u from LDS
        v2f a = *(const v2f*)(lrow + k);
        v2f b = *(const v2f*)(brow + HIDDEN + k);
        acc = wmma_f32_k4(a, b, acc);
    }

    const int n  = nt * 16 + r16;
    const int m0 = mt * 16 + 8 * khi;
    const float bn = bias[n];
    #pragma unroll
    for (int r = 0; r < 8; ++r) {
        const size_t off = (size_t)(m0 + r) * HIDDEN + n;
        out[off] = silu(acc[r] + bn) + EA[off];
    }
}

// ---------------------------------------------------------------------------
extern "C" void kernel_launch(void* const* d_in, const int* in_sizes, int n_in,
                              void* d_out, int out_size, void* d_ws, size_t ws_size,
                              hipStream_t stream) {
    const float* edge_attr = (const float*)d_in[0];
    const float* sbf       = (const float*)d_in[1];
    const int*   idx_kj    = (const int*)d_in[2];
    const int*   idx_ji    = (const int*)d_in[3];
    const float* W_sbf0    = (const float*)d_in[4];
    const float* W_sbf     = (const float*)d_in[5];
    const float* W_e1      = (const float*)d_in[6];
    const float* b_e1      = (const float*)d_in[7];
    const float* W_down    = (const float*)d_in[8];
    const float* W_up      = (const float*)d_in[9];
    float* out = (float*)d_out;

    // Workspace layout (floats): X[E*64] | AGG[E*64] | Wc[64*8]  (~128 MB)
    float* X   = (float*)d_ws;
    float* AGG = X   + (size_t)E_EDGES * INT_EMB;
    float* WC  = AGG + (size_t)E_EDGES * INT_EMB;

    k_prep<<<1, 512, 0, stream>>>(W_sbf0, W_sbf, WC);

    const int n4 = E_EDGES * INT_EMB / 4;
    k_zero<<<n4 / 256, 256, 0, stream>>>((float4*)AGG, n4);

    // 16384 M-tiles * 4 N-tiles, 8 waves/block
    k_gemm_down<<<MT_TILES * 4 / 8, 256, 0, stream>>>(edge_attr, W_down, X);

    // one wave per triplet, 8 waves/block
    k_triplet<<<T_TRIP / 8, 256, 0, stream>>>(sbf, idx_kj, idx_ji, WC, X, AGG);

    // fused up-projection + edge update: one M-tile per block
    k_up_e1<<<MT_TILES, 256, 0, stream>>>(AGG, W_up, edge_attr, W_e1, b_e1, out);
}